// MoDRouter_18356690223154
// MI455X (gfx1250) — compile-verified
//
#include <hip/hip_runtime.h>
#include <hip/hip_bf16.h>

typedef __attribute__((ext_vector_type(2))) float v2f;
typedef __attribute__((ext_vector_type(8))) float v8f;

#define NB   4
#define SS   8192
#define DD   2048
#define KSEL 4096   /* CAPACITY_FACTOR * S */

// ---------------------------------------------------------------------------
// Kernel 1: logits = hidden @ w + bias, probs = sigmoid(logits)
// One wave per 16-token tile. V_WMMA_F32_16X16X4_F32 accumulates over D.
//   A (16x4 f32): lane (row = lane&15, khi = lane>>4) holds
//                 h[row][k0 + 2*khi + {0,1}]  in a[0], a[1]
//   B (4x16 f32): same k mapping, weight replicated across all 16 columns:
//                 b[v] = w[k0 + 2*khi + v]   (independent of lane&15)
// => every column of D holds the logit; D layout: vgpr j, lane L ->
//    M = j + 8*(L>>4), N = L&15.
// ---------------------------------------------------------------------------
__global__ __launch_bounds__(256) void mod_router_logits_kernel(
    const float* __restrict__ hidden,   // [NB*SS, DD]
    const float* __restrict__ w,        // [DD]
    const float* __restrict__ bias,     // [1]
    float* __restrict__ probs)          // [NB*SS]
{
  const int lane = threadIdx.x & 31;
  const int wv   = threadIdx.x >> 5;
  const int tile = blockIdx.x * 8 + wv;        // 16 tokens per tile
  const int row  = lane & 15;
  const int khi  = lane >> 4;

  const float* hrow = hidden + (size_t)(tile * 16 + row) * DD + 2 * khi;
  const float* wrow = w + 2 * khi;

  v8f acc = {};
  #pragma unroll 4
  for (int k0 = 0; k0 < DD; k0 += 4) {
    v2f a = *(const v2f*)(hrow + k0);
    v2f b = *(const v2f*)(wrow + k0);
    // (neg_a, A, neg_b, B, c_mod, C, reuse_a, reuse_b)
    acc = __builtin_amdgcn_wmma_f32_16x16x4_f32(
        false, a, false, b, (short)0, acc, false, false);
  }

  const float bv = bias[0];
  #pragma unroll
  for (int j = 0; j < 8; ++j) {
    if (row == j) {                       // lanes j and j+16 -> tokens j, j+8
      float logit = acc[j] + bv;
      probs[tile * 16 + j + 8 * khi] = 1.0f / (1.0f + __expf(-logit));
    }
  }
}

// ---------------------------------------------------------------------------
// Kernel 2 helpers: block-wide reductions / scan (1024 threads = 32 waves)
// ---------------------------------------------------------------------------
__device__ __forceinline__ int blockReduceSumI(int v, int* red) {
  for (int off = 16; off; off >>= 1) v += __shfl_xor(v, off, 32);
  const int lane = threadIdx.x & 31, wv = threadIdx.x >> 5;
  if (lane == 0) red[wv] = v;
  __syncthreads();
  if (wv == 0) {
    int s = red[lane];
    for (int off = 16; off; off >>= 1) s += __shfl_xor(s, off, 32);
    if (lane == 0) red[0] = s;
  }
  __syncthreads();
  int r = red[0];
  __syncthreads();                       // protect red[] before next reuse
  return r;
}

__device__ __forceinline__ float blockReduceSumF(float v, float* red) {
  for (int off = 16; off; off >>= 1) v += __shfl_xor(v, off, 32);
  const int lane = threadIdx.x & 31, wv = threadIdx.x >> 5;
  if (lane == 0) red[wv] = v;
  __syncthreads();
  if (wv == 0) {
    float s = red[lane];
    for (int off = 16; off; off >>= 1) s += __shfl_xor(s, off, 32);
    if (lane == 0) red[0] = s;
  }
  __syncthreads();
  float r = red[0];
  __syncthreads();
  return r;
}

// ---------------------------------------------------------------------------
// Kernel 2: per batch row -> aux loss + exact top-k(4096) mask.
// Radix-select on float bits (all probs > 0, so uint order == float order),
// ties resolved in increasing index order (jax.lax.top_k semantics).
// ---------------------------------------------------------------------------
__global__ __launch_bounds__(1024) void mod_router_select_kernel(
    const float* __restrict__ probs,    // [NB*SS] (row b at b*SS)
    float* __restrict__ mask,           // [NB*SS]
    float* __restrict__ aux)            // [NB]
{
  __shared__ int   scn[1024];
  __shared__ int   redi[32];
  __shared__ float redf[32];

  const int b   = blockIdx.x;
  const int tid = threadIdx.x;
  const float* rowp = probs + b * SS;

  // 8 contiguous elements per thread, kept in registers
  float4 p0 = ((const float4*)rowp)[tid * 2 + 0];
  float4 p1 = ((const float4*)rowp)[tid * 2 + 1];
  float lp[8] = {p0.x, p0.y, p0.z, p0.w, p1.x, p1.y, p1.z, p1.w};
  unsigned lu[8];
  float lsum = 0.0f;
  #pragma unroll
  for (int i = 0; i < 8; ++i) { lu[i] = __float_as_uint(lp[i]); lsum += lp[i]; }

  // aux loss: 0.01 * (mean - 0.5)^2
  float tot = blockReduceSumF(lsum, redf);
  if (tid == 0) {
    float d = tot * (1.0f / (float)SS) - 0.5f;
    aux[b] = 0.01f * d * d;
  }

  // radix select: largest x with count(u >= x) >= KSEL  ==  k-th largest value
  unsigned x = 0u;
  for (int bit = 30; bit >= 0; --bit) {
    unsigned cand = x | (1u << bit);
    int c = 0;
    #pragma unroll
    for (int i = 0; i < 8; ++i) c += (lu[i] >= cand) ? 1 : 0;
    c = blockReduceSumI(c, redi);
    if (c >= KSEL) x = cand;             // uniform: c is block-broadcast
  }
  const unsigned tau = x;

  int cg = 0;
  #pragma unroll
  for (int i = 0; i < 8; ++i) cg += (lu[i] > tau) ? 1 : 0;
  cg = blockReduceSumI(cg, redi);
  const int r = KSEL - cg;               // ties to admit, lowest indices first

  // exclusive prefix of tie counts (Hillis-Steele over 1024 threads)
  int tie = 0;
  #pragma unroll
  for (int i = 0; i < 8; ++i) tie += (lu[i] == tau) ? 1 : 0;
  scn[tid] = tie;
  __syncthreads();
  for (int off = 1; off < 1024; off <<= 1) {
    int v = (tid >= off) ? scn[tid - off] : 0;
    __syncthreads();
    scn[tid] += v;
    __syncthreads();
  }
  int running = scn[tid] - tie;          // exclusive prefix

  float* mrow = mask + b * SS;
  #pragma unroll
  for (int i = 0; i < 8; ++i) {
    unsigned u = lu[i];
    float m;
    if (u > tau)       m = 1.0f;
    else if (u == tau) { m = (running < r) ? 1.0f : 0.0f; running++; }
    else               m = 0.0f;
    mrow[tid * 8 + i] = m;
  }
}

// ---------------------------------------------------------------------------
extern "C" void kernel_launch(void* const* d_in, const int* in_sizes, int n_in,
                              void* d_out, int out_size, void* d_ws, size_t ws_size,
                              hipStream_t stream) {
  const float* hidden = (const float*)d_in[0];   // [4*8192*2048] fp32
  const float* w      = (const float*)d_in[1];   // [2048] fp32
  const float* bias   = (const float*)d_in[2];   // [1] fp32

  float* out   = (float*)d_out;
  float* probs = out;                    // [32768]
  float* mask  = out + NB * SS;          // [32768]
  float* aux   = out + 2 * NB * SS;      // [4]

  const int tiles = (NB * SS) / 16;      // 2048 tiles -> 2048 waves
  mod_router_logits_kernel<<<tiles / 8, 256, 0, stream>>>(hidden, w, bias, probs);
  mod_router_select_kernel<<<NB, 1024, 0, stream>>>(probs, mask, aux);
}